// LRSSM_30597347017047
// MI455X (gfx1250) — compile-verified
//
#include <hip/hip_runtime.h>

// S4D / diagonal-SSM reservoir + 2-layer 1x1-conv MLP for MI455X (gfx1250).
// Shapes: B=16, H=128, L=4096, N=64.  dt = 0.01.

#define DT 0.01f
constexpr int Bb = 16;
constexpr int Hh = 128;
constexpr int Ll = 4096;
constexpr int Nn = 64;

typedef __attribute__((ext_vector_type(2))) float v2f;
typedef __attribute__((ext_vector_type(4))) float v4f;
typedef __attribute__((ext_vector_type(8))) float v8f;

// ---------------------------------------------------------------------------
// Kernel 1: diagonal SSM scan.
//   s_n[l] = A_n * s_n[l-1] + u[l],  A_n = exp(dt*Lambda_n)
//   y[b,h,l] = sum_n Re(W_n * s_n[l]) + d[h]*u[b,h,l]
// One wave32 per (b,h) sequence; 2 complex states per lane (N=64).
// 32-step chunks; per-lane partial readouts transposed through LDS so the
// final y store is coalesced.
// ---------------------------------------------------------------------------
__global__ __launch_bounds__(256) void ssm_scan_kernel(
    const float* __restrict__ u,   const float* __restrict__ lre,
    const float* __restrict__ lim, const float* __restrict__ wre,
    const float* __restrict__ wim, const float* __restrict__ dvec,
    float* __restrict__ y)
{
    __shared__ float slab[8][32 * 33];   // per-wave 32x32 slab, padded stride 33

    const int lane = threadIdx.x & 31;
    const int wid  = threadIdx.x >> 5;
    const int seq  = blockIdx.x * 8 + wid;      // flat (b*H + h)
    const int h    = seq & (Hh - 1);

    const float* up = u + (size_t)seq * Ll;
    float*       yp = y + (size_t)seq * Ll;

    // two states per lane: n and n+32
    const int n0 = lane, n1 = lane + 32;
    const float lr0 = lre[h * Nn + n0], li0 = lim[h * Nn + n0];
    const float lr1 = lre[h * Nn + n1], li1 = lim[h * Nn + n1];
    const float e0  = __expf(DT * lr0), e1 = __expf(DT * lr1);
    const float ar0 = e0 * __cosf(DT * li0), ai0 = e0 * __sinf(DT * li0);
    const float ar1 = e1 * __cosf(DT * li1), ai1 = e1 * __sinf(DT * li1);
    const float wr0 = wre[h * Nn + n0], wi0 = wim[h * Nn + n0];
    const float wr1 = wre[h * Nn + n1], wi1 = wim[h * Nn + n1];
    const float dh  = dvec[h];

    float sr0 = 0.f, si0 = 0.f, sr1 = 0.f, si1 = 0.f;
    float* my = slab[wid];

    for (int l0 = 0; l0 < Ll; l0 += 32) {
        const float uc = up[l0 + lane];          // coalesced 32-wide chunk
        #pragma unroll
        for (int t = 0; t < 32; ++t) {
            const float ut = __shfl(uc, t, 32);  // broadcast u[l0+t]
            const float nr0 = fmaf(ar0, sr0, fmaf(-ai0, si0, ut));
            const float ni0 = fmaf(ar0, si0, ai0 * sr0);
            const float nr1 = fmaf(ar1, sr1, fmaf(-ai1, si1, ut));
            const float ni1 = fmaf(ar1, si1, ai1 * sr1);
            sr0 = nr0; si0 = ni0; sr1 = nr1; si1 = ni1;
            // per-lane readout partial (2 of the 64 states)
            const float p = wr0 * sr0 - wi0 * si0 + wr1 * sr1 - wi1 * si1;
            my[t * 33 + lane] = p;
        }
        __syncthreads();
        float acc = 0.f;
        #pragma unroll
        for (int i = 0; i < 32; ++i) acc += my[lane * 33 + i];
        yp[l0 + lane] = acc + dh * uc;
        __syncthreads();
    }
}

// ---------------------------------------------------------------------------
// Kernels 2/3: out[b,o,l] = relu( sum_h W[o,h] * X[b,h,l] + bias[o] )
// GEMM: (128x128) @ (128 x 65536) with V_WMMA_F32_16X16X4_F32.
// Block = 256 threads (8 waves); block owns 128 contiguous columns (b,l),
// wave w owns 16 columns and all 8 M-tiles (8 x v8f accumulators).
// W is staged in LDS; A-tiles come back via ds_load_b64 in the ISA's
// 16x4 f32 A layout (lanes 0-15: K={0,1}; lanes 16-31: K={2,3}).
// ---------------------------------------------------------------------------
__global__ __launch_bounds__(256) void mlp_wmma_kernel(
    const float* __restrict__ W, const float* __restrict__ bias,
    const float* __restrict__ X, float* __restrict__ Y)
{
    __shared__ float sW[Hh * Hh];                 // 64 KB
    for (int i = threadIdx.x; i < (Hh * Hh) / 4; i += 256)
        ((v4f*)sW)[i] = ((const v4f*)W)[i];
    __syncthreads();

    const int lane = threadIdx.x & 31;
    const int wid  = threadIdx.x >> 5;
    const int n    = lane & 15;                   // column within 16-wide tile
    const int hi   = lane >> 4;                   // half-wave selector

    const long c0  = (long)blockIdx.x * 128 + wid * 16;  // flat (b*L + l)
    const int  bb  = (int)(c0 / Ll);
    const int  l0  = (int)(c0 % Ll);              // L % 128 == 0: b uniform

    const float* xcol = X + ((size_t)bb * Hh) * Ll + l0 + n;

    v8f acc[8] = {};                              // 8 M-tiles x 16x16 f32

    for (int kt = 0; kt < 32; ++kt) {
        const int k0 = kt * 4;
        // B tile 4x16: lane(0-15) rows K={0,1}, lane(16-31) rows K={2,3}
        const int rb = k0 + 2 * hi;
        v2f bv;
        bv.x = xcol[(size_t)rb * Ll];
        bv.y = xcol[(size_t)(rb + 1) * Ll];
        #pragma unroll
        for (int mt = 0; mt < 8; ++mt) {
            // A tile 16x4: lane m -> {W[o0+m,k0+2hi], W[o0+m,k0+2hi+1]}
            const int row = mt * 16 + n;
            const v2f av = *(const v2f*)&sW[row * Hh + k0 + 2 * hi];
            acc[mt] = __builtin_amdgcn_wmma_f32_16x16x4_f32(
                /*neg_a=*/false, av, /*neg_b=*/false, bv,
                /*c_mod=*/(short)0, acc[mt],
                /*reuse_a=*/false, /*reuse_b=*/false);
        }
    }

    // C/D layout: VGPR v, lanes 0-15 -> row v, lanes 16-31 -> row v+8
    float* ycol = Y + ((size_t)bb * Hh) * Ll + l0 + n;
    #pragma unroll
    for (int mt = 0; mt < 8; ++mt) {
        #pragma unroll
        for (int v = 0; v < 8; ++v) {
            const int row = mt * 16 + v + 8 * hi;
            float r = acc[mt][v] + bias[row];
            ycol[(size_t)row * Ll] = r > 0.f ? r : 0.f;
        }
    }
}

// ---------------------------------------------------------------------------
// Launch. Inputs in setup_inputs() order:
//  0:u 1:lambda_re 2:lambda_im 3:w_re 4:w_im 5:d 6:mlp_w1 7:mlp_b1 8:mlp_w2 9:mlp_b2
// d_ws holds y (B*H*L f32 = 32 MB). Layer-1 output goes to d_out, layer-2
// runs safely in place on d_out (column-disjoint blocks, reads before writes).
// ---------------------------------------------------------------------------
extern "C" void kernel_launch(void* const* d_in, const int* in_sizes, int n_in,
                              void* d_out, int out_size, void* d_ws, size_t ws_size,
                              hipStream_t stream)
{
    const float* u   = (const float*)d_in[0];
    const float* lre = (const float*)d_in[1];
    const float* lim = (const float*)d_in[2];
    const float* wre = (const float*)d_in[3];
    const float* wim = (const float*)d_in[4];
    const float* dv  = (const float*)d_in[5];
    const float* w1  = (const float*)d_in[6];
    const float* b1  = (const float*)d_in[7];
    const float* w2  = (const float*)d_in[8];
    const float* b2  = (const float*)d_in[9];
    float* out = (float*)d_out;
    float* yws = (float*)d_ws;                    // B*H*L floats

    // 1) SSM scan: 2048 sequences, 8 waves (sequences) per 256-thread block
    ssm_scan_kernel<<<(Bb * Hh) / 8, 256, 0, stream>>>(u, lre, lim, wre, wim, dv, yws);

    // 2) MLP layer 1: h1 = relu(W1 @ y + b1) -> d_out
    const int nblk = (Bb * Ll) / 128;             // 512 column-blocks
    mlp_wmma_kernel<<<nblk, 256, 0, stream>>>(w1, b1, yws, out);

    // 3) MLP layer 2: out = relu(W2 @ h1 + b2), in place on d_out
    mlp_wmma_kernel<<<nblk, 256, 0, stream>>>(w2, b2, out, out);
}